// SimpleGNN_30580167147629
// MI455X (gfx1250) — compile-verified
//
#include <hip/hip_runtime.h>

typedef float v2f __attribute__((ext_vector_type(2)));
typedef float v8f __attribute__((ext_vector_type(8)));

#define DIM 64

// ---------------------------------------------------------------------------
// Degree / normalization kernels
// ---------------------------------------------------------------------------
__global__ void gcn_deg_init(float* deg, int n) {
    int i = blockIdx.x * blockDim.x + threadIdx.x;
    if (i < n) deg[i] = 1.0f;   // self-loop contributes 1
}

__global__ void gcn_deg_accum(const int* __restrict__ dst, float* deg, int E) {
    int e = blockIdx.x * blockDim.x + threadIdx.x;
    if (e < E) unsafeAtomicAdd(&deg[dst[e]], 1.0f);
}

__global__ void gcn_deg_to_dinv(float* deg, int n) {
    int i = blockIdx.x * blockDim.x + threadIdx.x;
    if (i < n) {
        float d = deg[i];
        deg[i] = (d > 0.0f) ? rsqrtf(d) : 0.0f;
    }
}

// ---------------------------------------------------------------------------
// Dense GEMM: T = act(A) @ W     (A: nRows x 64, W: 64 x 64, T: nRows x 64)
// One wave computes a 16-row x 64-col tile using V_WMMA_F32_16X16X4_F32.
// A-frag (16x4 f32, ISA 7.12.2): lanes 0-15 -> K={k,k+1}, lanes 16-31 -> K={k+2,k+3}
// B-frag (4x16 f32): lane holds one output column, same K striping.
// C/D (16x16 f32, 8 VGPRs): lanes 0-15 rows 0..7, lanes 16-31 rows 8..15.
// ---------------------------------------------------------------------------
template <int RELU>
__global__ __launch_bounds__(256) void gcn_gemm_wmma(const float* __restrict__ A,
                                                     const float* __restrict__ W,
                                                     float* __restrict__ T, int nRows) {
    const int wave = threadIdx.x >> 5;
    const int lane = threadIdx.x & 31;
    const int tile = blockIdx.x * (blockDim.x >> 5) + wave;
    const int r0   = tile * 16;
    if (r0 >= nRows) return;               // wave-uniform guard (EXEC stays all-1s)

    const int half = lane >> 4;            // 0: K pair {k,k+1}, 1: {k+2,k+3}
    const int l    = lane & 15;
    const size_t rowA = (size_t)(r0 + l) * DIM;

    v8f acc0 = {}, acc1 = {}, acc2 = {}, acc3 = {};

    #pragma unroll
    for (int k = 0; k < DIM; k += 4) {
        const int ka = k + half * 2;
        v2f a = *(const v2f*)(A + rowA + ka);   // contiguous K pair, 8B aligned
        if (RELU) { a.x = fmaxf(a.x, 0.0f); a.y = fmaxf(a.y, 0.0f); }

        const float* Wr0 = W + (size_t)ka * DIM;
        const float* Wr1 = Wr0 + DIM;
        v2f b0, b1, b2, b3;
        b0.x = Wr0[l +  0]; b0.y = Wr1[l +  0];
        b1.x = Wr0[l + 16]; b1.y = Wr1[l + 16];
        b2.x = Wr0[l + 32]; b2.y = Wr1[l + 32];
        b3.x = Wr0[l + 48]; b3.y = Wr1[l + 48];

        acc0 = __builtin_amdgcn_wmma_f32_16x16x4_f32(false, a, false, b0, (short)0, acc0, false, false);
        acc1 = __builtin_amdgcn_wmma_f32_16x16x4_f32(false, a, false, b1, (short)0, acc1, false, false);
        acc2 = __builtin_amdgcn_wmma_f32_16x16x4_f32(false, a, false, b2, (short)0, acc2, false, false);
        acc3 = __builtin_amdgcn_wmma_f32_16x16x4_f32(false, a, false, b3, (short)0, acc3, false, false);
    }

    // store D tiles: lane half selects rows 0..7 vs 8..15 of the 16-row tile
    float* out = T + (size_t)(r0 + half * 8) * DIM + l;
    #pragma unroll
    for (int r = 0; r < 8; ++r) {
        out[(size_t)r * DIM +  0] = acc0[r];
        out[(size_t)r * DIM + 16] = acc1[r];
        out[(size_t)r * DIM + 32] = acc2[r];
        out[(size_t)r * DIM + 48] = acc3[r];
    }
}

// ---------------------------------------------------------------------------
// Aggregation: out[i][:] = xw[i][:]*dinv[i]^2 + b   (self-loop + bias init)
// ---------------------------------------------------------------------------
__global__ void gcn_agg_init(const float* __restrict__ xw, const float* __restrict__ dinv,
                             const float* __restrict__ b, float* __restrict__ out, int n) {
    int i = blockIdx.x * blockDim.x + threadIdx.x;
    if (i >= n) return;
    int node = i >> 6;
    int col  = i & 63;
    float di = dinv[node];
    out[i] = xw[i] * di * di + b[col];
}

// One wave per edge: lanes cover the 64-float feature row (2 floats/lane).
__global__ __launch_bounds__(256) void gcn_edge_scatter(const float* __restrict__ xw,
                                                        const int* __restrict__ src,
                                                        const int* __restrict__ dst,
                                                        const float* __restrict__ dinv,
                                                        float* out, int E) {
    int wave = (blockIdx.x * blockDim.x + threadIdx.x) >> 5;
    int lane = threadIdx.x & 31;
    if (wave >= E) return;
    int s = src[wave];
    int d = dst[wave];
    float nrm = dinv[s] * dinv[d];
    const float* row = xw + (size_t)s * DIM;
    float* orow = out + (size_t)d * DIM;
    unsafeAtomicAdd(&orow[lane],      row[lane]      * nrm);
    unsafeAtomicAdd(&orow[lane + 32], row[lane + 32] * nrm);
}

__global__ void gcn_relu(float* x, int n) {
    int i = blockIdx.x * blockDim.x + threadIdx.x;
    if (i < n) x[i] = fmaxf(x[i], 0.0f);
}

// ---------------------------------------------------------------------------
// Launch
// ---------------------------------------------------------------------------
extern "C" void kernel_launch(void* const* d_in, const int* in_sizes, int n_in,
                              void* d_out, int out_size, void* d_ws, size_t ws_size,
                              hipStream_t stream) {
    const float* x   = (const float*)d_in[0];
    const int*   ei  = (const int*)d_in[1];      // (2, E): first E = src, next E = dst
    const float* W1  = (const float*)d_in[2];
    const float* b1  = (const float*)d_in[3];
    const float* W2  = (const float*)d_in[4];
    const float* b2  = (const float*)d_in[5];
    float*       out = (float*)d_out;

    const int N = in_sizes[0] / DIM;
    const int E = in_sizes[1] / 2;
    const int* src = ei;
    const int* dst = ei + E;

    float* ws   = (float*)d_ws;
    float* dinv = ws;                                  // N floats (deg, then dinv)
    size_t off  = ((size_t)N + 255) & ~(size_t)255;
    float* t0   = ws + off;                            // N*64: xw of current layer
    float* t1   = t0 + (size_t)N * DIM;                // N*64: aggregated h1 (pre-relu)

    const int T256 = 256;
    const int nblk_N  = (N + T256 - 1) / T256;
    const int nblk_E  = (E + T256 - 1) / T256;
    const int nblk_NF = (N * DIM + T256 - 1) / T256;
    const int tiles   = (N + 15) / 16;
    const int nblk_G  = (tiles + 7) / 8;               // 8 waves (tiles) per block
    const int nblk_S  = (E + 7) / 8;                   // 8 edges (waves) per block

    // normalization: deg -> dinv
    gcn_deg_init<<<nblk_N, T256, 0, stream>>>(dinv, N);
    gcn_deg_accum<<<nblk_E, T256, 0, stream>>>(dst, dinv, E);
    gcn_deg_to_dinv<<<nblk_N, T256, 0, stream>>>(dinv, N);

    // layer 1: t0 = x @ W1 ; t1 = scatter(t0) + b1   (relu deferred)
    gcn_gemm_wmma<0><<<nblk_G, T256, 0, stream>>>(x, W1, t0, N);
    gcn_agg_init<<<nblk_NF, T256, 0, stream>>>(t0, dinv, b1, t1, N * DIM);
    gcn_edge_scatter<<<nblk_S, T256, 0, stream>>>(t0, src, dst, dinv, t1, E);

    // layer 2: t0 = relu(t1) @ W2 ; out = scatter(t0) + b2 ; relu(out)
    gcn_gemm_wmma<1><<<nblk_G, T256, 0, stream>>>(t1, W2, t0, N);
    gcn_agg_init<<<nblk_NF, T256, 0, stream>>>(t0, dinv, b2, out, N * DIM);
    gcn_edge_scatter<<<nblk_S, T256, 0, stream>>>(t0, src, dst, dinv, out, E);
    gcn_relu<<<nblk_NF, T256, 0, stream>>>(out, N * DIM);
}